// QuantumKernel_87333864997077
// MI455X (gfx1250) — compile-verified
//
#include <hip/hip_runtime.h>
#include <hip/hip_bf16.h>
#include <math.h>

// ---------------------------------------------------------------------------
// 9-qubit statevector simulator, one wave32 per sample, state (float2) in LDS.
// Embedding: closed-form product state. Layers: fused 16x16 complex unitaries
// applied with chained V_WMMA_F32_16X16X4_F32. CNOT rings: in-wave LDS perms.
// ---------------------------------------------------------------------------

typedef float v2f __attribute__((ext_vector_type(2)));
typedef float v8f __attribute__((ext_vector_type(8)));

// padded LDS index (complex-element units): +1 every 16 kills the stride-32
// (mid-group) bank conflicts; strided lanes land on banks 4*h mod 64.
#define LPAD(i) ((i) + ((i) >> 4))
// Compiler reorder barrier only. Each sample is owned by ONE wave and LDS ops
// of a wave are processed in order (DScnt, ISA 7.3), so cross-lane RAW through
// LDS needs only issue-order preservation -- no hardware drain required.
#define LDS_BAR() asm volatile("" ::: "memory")

__device__ __forceinline__ v8f wmma4(v2f a, v2f b, v8f c) {
  // D = A(16x4,f32) x B(4x16,f32) + C(16x16,f32)
  return __builtin_amdgcn_wmma_f32_16x16x4_f32(
      /*neg_a=*/false, a, /*neg_b=*/false, b,
      /*c_mod=*/(short)0, c, /*reuse_a=*/false, /*reuse_b=*/false);
}

struct C2 { float re, im; };
__device__ __forceinline__ C2 cmul(C2 a, C2 b) {
  return { a.re * b.re - a.im * b.im, a.re * b.im + a.im * b.re };
}

// ---------------------------------------------------------------------------
// Kernel 0: build fused layer unitaries from weights.
//  ws[0    ..1023] : gHi [layer][16*16][re,im]  (wires 0-3, Kronecker product)
//  ws[1024 ..2047] : gMid[layer][16*16][re,im]  (wires 4-7)
//  ws[2048 ..2063] : g8  [layer][2*2][re,im]    (wire 8)
// grid: 2 blocks (one per layer) x 256 threads (one per (r,c) entry)
// ---------------------------------------------------------------------------
__global__ void build_gates(const float* __restrict__ w, float* __restrict__ ws) {
  const int l   = blockIdx.x;
  const int tid = threadIdx.x;            // 0..255 -> (r,c) of the 16x16 gate

  C2 g[9][2][2];
  #pragma unroll
  for (int q = 0; q < 9; ++q) {
    const float phi   = w[(l * 9 + q) * 3 + 0];
    const float theta = w[(l * 9 + q) * 3 + 1];
    const float omega = w[(l * 9 + q) * 3 + 2];
    const float ct = __cosf(0.5f * theta), st = __sinf(0.5f * theta);
    const float app = 0.5f * (phi + omega), apm = 0.5f * (phi - omega);
    const C2 ep = { __cosf(app), -__sinf(app) };  // exp(-0.5i(phi+omega))
    const C2 em = { __cosf(apm),  __sinf(apm) };  // exp(+0.5i(phi-omega))
    g[q][0][0] = {  ep.re * ct,  ep.im * ct };    //  ep*ct
    g[q][0][1] = { -em.re * st, -em.im * st };    // -em*st
    g[q][1][0] = {  em.re * st, -em.im * st };    //  conj(em)*st
    g[q][1][1] = {  ep.re * ct, -ep.im * ct };    //  conj(ep)*ct
  }

  const int r = tid >> 4, c = tid & 15;
  C2 hi  = { 1.0f, 0.0f };
  C2 mid = { 1.0f, 0.0f };
  #pragma unroll
  for (int k = 0; k < 4; ++k) {           // wire k -> bit (3-k) of r/c
    const int rb = (r >> (3 - k)) & 1, cb = (c >> (3 - k)) & 1;
    hi  = cmul(hi,  g[k][rb][cb]);
    mid = cmul(mid, g[4 + k][rb][cb]);
  }
  ws[(l * 256 + tid) * 2 + 0]        = hi.re;
  ws[(l * 256 + tid) * 2 + 1]        = hi.im;
  ws[1024 + (l * 256 + tid) * 2 + 0] = mid.re;
  ws[1024 + (l * 256 + tid) * 2 + 1] = mid.im;
  if (tid < 4) {
    const C2 e = g[8][tid >> 1][tid & 1];
    ws[2048 + (l * 4 + tid) * 2 + 0] = e.re;
    ws[2048 + (l * 4 + tid) * 2 + 1] = e.im;
  }
}

// ---------------------------------------------------------------------------
// Kernel 1: main simulator. 256 threads = 8 waves per block, 1 sample / wave.
// ---------------------------------------------------------------------------
__global__ __launch_bounds__(256) void qsim(const float* __restrict__ x,
                                            const float* __restrict__ gates,
                                            float* __restrict__ out) {
  __shared__ float2 smem[8 * 544];        // per wave: 512 complex + padding
  const int wv   = threadIdx.x >> 5;
  const int lane = threadIdx.x & 31;
  float2* S = &smem[wv * 544];

  const int n   = blockIdx.x * 8 + wv;    // sample id, 0..32767
  const int b   = n >> 12;
  const int rem = n & 4095;
  const int ii  = rem >> 6;
  const int jj  = rem & 63;

  // ---- RX embedding angles (zero-padded 3x3 patch; angle 0 -> identity RX)
  float cq[9], sq[9];
  #pragma unroll
  for (int q = 0; q < 9; ++q) {
    const int ki = ii + q / 3 - 1;
    const int kj = jj + q % 3 - 1;
    float a = 0.0f;
    if (ki >= 0 && ki < 64 && kj >= 0 && kj < 64) a = x[b * 4096 + ki * 64 + kj];
    cq[q] = __cosf(0.5f * a);
    sq[q] = __sinf(0.5f * a);
  }

  // ---- product state: amp(i) = (-i)^popc(i) * prod_q (bit_q ? s_q : c_q)
  // wire q lives at bit (8-q); wire 0 = MSB (PennyLane convention)
  #pragma unroll
  for (int k = 0; k < 16; ++k) {
    const int i = k * 32 + lane;
    float mag = 1.0f;
    #pragma unroll
    for (int q = 0; q < 9; ++q) mag *= ((i >> (8 - q)) & 1) ? sq[q] : cq[q];
    const int p = __popc(i) & 3;
    const float re = (p == 0) ? mag : ((p == 2) ? -mag : 0.0f);
    const float im = (p == 1) ? -mag : ((p == 3) ? mag : 0.0f);
    S[LPAD(i)] = make_float2(re, im);
  }
  LDS_BAR();

  const int lhalf = lane >> 4;            // 0 | 1
  const int lmod  = lane & 15;

  for (int l = 0; l < 2; ++l) {
    // ================= hi group: 16x16 gate on bits 8:5 (wires 0-3) ========
    {
      const float* G = gates + l * 512;   // gHi[layer]
      v2f Ar[4], Ai[4], Ain[4];           // A frags per K-block (Gr, Gi, -Gi)
      #pragma unroll
      for (int kb = 0; kb < 4; ++kb)
        #pragma unroll
        for (int j = 0; j < 2; ++j) {
          const int K = 4 * kb + j + 2 * lhalf;
          const float gr = G[(lmod * 16 + K) * 2 + 0];
          const float gi = G[(lmod * 16 + K) * 2 + 1];
          Ar[kb][j] = gr; Ai[kb][j] = gi; Ain[kb][j] = -gi;
        }
      #pragma unroll
      for (int ct = 0; ct < 2; ++ct) {    // two 16-column tiles of the state
        v2f Br[4], Bi[4];
        #pragma unroll
        for (int kb = 0; kb < 4; ++kb)
          #pragma unroll
          for (int j = 0; j < 2; ++j) {
            const int K = 4 * kb + j + 2 * lhalf;
            const float2 v = S[LPAD(K * 32 + ct * 16 + lmod)];
            Br[kb][j] = v.x;
            Bi[kb][j] = v.y;
          }
        v8f aR = {0, 0, 0, 0, 0, 0, 0, 0};
        v8f aI = {0, 0, 0, 0, 0, 0, 0, 0};
        #pragma unroll
        for (int kb = 0; kb < 4; ++kb) {  // complex GEMM: 4 real products
          aR = wmma4(Ar[kb],  Br[kb], aR);
          aR = wmma4(Ain[kb], Bi[kb], aR);
          aI = wmma4(Ar[kb],  Bi[kb], aI);
          aI = wmma4(Ai[kb],  Br[kb], aI);
        }
        #pragma unroll
        for (int r = 0; r < 8; ++r) {     // D: (M = r+8*lhalf, N = lmod)
          const int didx = (r + 8 * lhalf) * 32 + ct * 16 + lmod;
          S[LPAD(didx)] = make_float2(aR[r], aI[r]);
        }
        LDS_BAR();
      }
    }

    // ================= mid group: 16x16 gate on bits 4:1 (wires 4-7) =======
    {
      const float* G = gates + 1024 + l * 512;  // gMid[layer]
      v2f Ar[4], Ai[4], Ain[4];
      #pragma unroll
      for (int kb = 0; kb < 4; ++kb)
        #pragma unroll
        for (int j = 0; j < 2; ++j) {
          const int K = 4 * kb + j + 2 * lhalf;
          const float gr = G[(lmod * 16 + K) * 2 + 0];
          const float gi = G[(lmod * 16 + K) * 2 + 1];
          Ar[kb][j] = gr; Ai[kb][j] = gi; Ain[kb][j] = -gi;
        }
      #pragma unroll
      for (int lp = 0; lp < 2; ++lp) {    // bit-0 parity selects the tile
        v2f Br[4], Bi[4];
        #pragma unroll
        for (int kb = 0; kb < 4; ++kb)
          #pragma unroll
          for (int j = 0; j < 2; ++j) {
            const int K = 4 * kb + j + 2 * lhalf;       // contraction = mid4
            const float2 v = S[LPAD(lmod * 32 + K * 2 + lp)];  // N = hi4
            Br[kb][j] = v.x;
            Bi[kb][j] = v.y;
          }
        v8f aR = {0, 0, 0, 0, 0, 0, 0, 0};
        v8f aI = {0, 0, 0, 0, 0, 0, 0, 0};
        #pragma unroll
        for (int kb = 0; kb < 4; ++kb) {
          aR = wmma4(Ar[kb],  Br[kb], aR);
          aR = wmma4(Ain[kb], Bi[kb], aR);
          aI = wmma4(Ar[kb],  Bi[kb], aI);
          aI = wmma4(Ai[kb],  Br[kb], aI);
        }
        #pragma unroll
        for (int r = 0; r < 8; ++r) {     // D: (M = new mid4, N = hi4)
          const int didx = lmod * 32 + (r + 8 * lhalf) * 2 + lp;
          S[LPAD(didx)] = make_float2(aR[r], aI[r]);
        }
        LDS_BAR();
      }
    }

    // ================= wire 8 gate: 2x2 on bit 0, pure VALU ================
    {
      const float* g8 = gates + 2048 + l * 8;
      const float a00r = g8[0], a00i = g8[1], a01r = g8[2], a01i = g8[3];
      const float a10r = g8[4], a10i = g8[5], a11r = g8[6], a11i = g8[7];
      #pragma unroll
      for (int k = 0; k < 8; ++k) {       // 256 pairs / 32 lanes
        const int p  = k * 32 + lane;
        const int i0 = 2 * p, i1 = 2 * p + 1;
        const float2 s0 = S[LPAD(i0)];
        const float2 s1 = S[LPAD(i1)];
        const float n0r = a00r * s0.x - a00i * s0.y + a01r * s1.x - a01i * s1.y;
        const float n0i = a00r * s0.y + a00i * s0.x + a01r * s1.y + a01i * s1.x;
        const float n1r = a10r * s0.x - a10i * s0.y + a11r * s1.x - a11i * s1.y;
        const float n1i = a10r * s0.y + a10i * s0.x + a11r * s1.y + a11i * s1.x;
        S[LPAD(i0)] = make_float2(n0r, n0i);
        S[LPAD(i1)] = make_float2(n1r, n1i);
      }
      LDS_BAR();
    }

    // ========== CNOT ring, range r = l+1: composed index permutation =======
    // reference applies perm_q for q=0..8 sequentially: total P = p0∘p1∘…∘p8
    {
      const int rr = l + 1;               // l % 8 + 1 for l in {0,1}
      float2 t[16];
      #pragma unroll
      for (int k = 0; k < 16; ++k) {
        const int i = k * 32 + lane;
        int j = i;
        #pragma unroll
        for (int q = 8; q >= 0; --q) {    // innermost perm first
          const int cb = 8 - q;
          const int tb = 8 - ((q + rr) % 9);
          j ^= ((j >> cb) & 1) << tb;
        }
        t[k] = S[LPAD(j)];
      }
      LDS_BAR();                          // all gathers issued before scatters
      #pragma unroll
      for (int k = 0; k < 16; ++k) {
        const int i = k * 32 + lane;
        S[LPAD(i)] = t[k];
      }
      LDS_BAR();
    }
  }

  // ---- <Z_0> = sum |amp(i)|^2 * (bit8(i) ? -1 : +1), wave-local reduction
  float pz = 0.0f;
  #pragma unroll
  for (int k = 0; k < 16; ++k) {
    const int i = k * 32 + lane;
    const float2 s = S[LPAD(i)];
    const float v = s.x * s.x + s.y * s.y;
    pz += (i & 256) ? -v : v;
  }
  #pragma unroll
  for (int off = 16; off > 0; off >>= 1) pz += __shfl_down(pz, off, 32);
  if (lane == 0) out[n] = pz;
}

// ---------------------------------------------------------------------------
extern "C" void kernel_launch(void* const* d_in, const int* in_sizes, int n_in,
                              void* d_out, int out_size, void* d_ws, size_t ws_size,
                              hipStream_t stream) {
  const float* x = (const float*)d_in[0];   // (8,1,64,64) f32
  const float* w = (const float*)d_in[1];   // (2,9,3) f32
  float* out = (float*)d_out;               // (8,1,64,64) f32
  float* ws  = (float*)d_ws;                // needs 2064 floats

  hipLaunchKernelGGL(build_gates, dim3(2), dim3(256), 0, stream, w, ws);
  hipLaunchKernelGGL(qsim, dim3(32768 / 8), dim3(256), 0, stream, x, ws, out);
}